// GCN_4698694222079
// MI455X (gfx1250) — compile-verified
//
#include <hip/hip_runtime.h>
#include <hip/hip_bf16.h>
#include <cstdint>

// ---------------------------------------------------------------------------
// GCN layer: h = X@W (f32 WMMA), agg = COO-SpMM(A, h) via f32 global atomics,
// out = selu(h*skip + agg + bias)
// ---------------------------------------------------------------------------

constexpr int D_IN  = 256;   // input feature dim
constexpr int C_OUT = 128;   // output channel dim

// GEMM blocking: 64 rows (4 M-tiles of 16) per block, 8 waves cover N=128.
constexpr int MT_PER_BLOCK = 4;
constexpr int ROWS_PER_BLOCK = 16 * MT_PER_BLOCK;       // 64
constexpr int LDS_STRIDE = D_IN + 4;                    // 260 floats: bank-conflict-free
                                                        // (row stride 1040B, 16B aligned)

typedef __attribute__((ext_vector_type(2))) float v2f;
typedef __attribute__((ext_vector_type(8))) float v8f;

// ---------------------------------------------------------------------------
// Kernel 1: zero the output/accumulator buffer (harness poisons d_out)
// ---------------------------------------------------------------------------
__global__ __launch_bounds__(256) void gcn_zero_kernel(float* __restrict__ out,
                                                       int total) {
    int idx = blockIdx.x * 256 + threadIdx.x;
    if (idx < total) out[idx] = 0.0f;
}

// ---------------------------------------------------------------------------
// Kernel 2: h = X @ W using V_WMMA_F32_16X16X4_F32.
// Block = 256 threads = 8 waves. Block b computes rows [64b, 64b+64) x 128
// cols. Wave w owns N-tile w (cols 16w..16w+15) and 4 M-tiles; B fragments
// are loaded once per k-step and reused across the 4 WMMAs.
//
// A fragment (16x4 f32, ISA layout): lane L<16 -> (M=L, K=khalf+{0,1}),
// lane L>=16 -> (M=L-16, K=2+{0,1}).
// B fragment (4x16): lane L -> (K=khalf+{0,1}, N=w*16+(L&15)).
// C/D (16x16 f32): VGPR r, lane L -> (M = r + (L>=16 ? 8 : 0), N = L&15).
// ---------------------------------------------------------------------------
__global__ __launch_bounds__(256) void gcn_gemm_wmma_kernel(
    const float* __restrict__ X,   // [n, 256]
    const float* __restrict__ W,   // [256, 128]
    float* __restrict__ H,         // [n, 128]
    int n) {
    __shared__ float As[ROWS_PER_BLOCK * LDS_STRIDE];   // 65 KB A slab (padded)

    const int tid  = threadIdx.x;
    const int row0 = blockIdx.x * ROWS_PER_BLOCK;

    // Cooperative, coalesced staging: 64 rows x 64 float4 per row.
    // Zero-pad rows beyond n (M tail).
    for (int i = tid; i < ROWS_PER_BLOCK * (D_IN / 4); i += 256) {
        int r  = i >> 6;                 // 0..63
        int c4 = i & 63;                 // float4 index within row
        float4 v;
        if (row0 + r < n) {
            v = reinterpret_cast<const float4*>(X + (size_t)(row0 + r) * D_IN)[c4];
        } else {
            v = make_float4(0.f, 0.f, 0.f, 0.f);
        }
        *reinterpret_cast<float4*>(&As[r * LDS_STRIDE + c4 * 4]) = v;
    }
    __syncthreads();

    const int wave  = tid >> 5;                     // 0..7 -> N tile
    const int lane  = tid & 31;
    const int l16   = lane & 15;
    const int khalf = (lane >> 4) << 1;             // 0 or 2
    const int col   = wave * 16 + l16;

    v8f acc[MT_PER_BLOCK];
#pragma unroll
    for (int mt = 0; mt < MT_PER_BLOCK; ++mt)
        acc[mt] = v8f{0.f, 0.f, 0.f, 0.f, 0.f, 0.f, 0.f, 0.f};

    const float* Wc = W + col;                      // this wave's B column

#pragma unroll 4
    for (int k0 = 0; k0 < D_IN; k0 += 4) {
        // B fragment: two dwords (rows k0+khalf, k0+khalf+1), shared by 4 WMMAs
        v2f b;
        b.x = Wc[(k0 + khalf) * C_OUT];
        b.y = Wc[(k0 + khalf + 1) * C_OUT];
        // 4 M-tiles: A fragment from LDS (8B-aligned, bank-conflict-free)
#pragma unroll
        for (int mt = 0; mt < MT_PER_BLOCK; ++mt) {
            v2f a = *reinterpret_cast<const v2f*>(
                &As[(mt * 16 + l16) * LDS_STRIDE + k0 + khalf]);
            acc[mt] = __builtin_amdgcn_wmma_f32_16x16x4_f32(
                false, a, false, b, (short)0, acc[mt], false, false);
        }
    }

    // Store: fast unguarded path for full tiles, guarded path only at M tail
    const int mofs = (lane >> 4) << 3;              // 0 or 8
#pragma unroll
    for (int mt = 0; mt < MT_PER_BLOCK; ++mt) {
        const int rowbase = row0 + mt * 16;
        if (rowbase + 15 < n) {
            float* dst = H + (size_t)(rowbase + mofs) * C_OUT + col;
#pragma unroll
            for (int r = 0; r < 8; ++r)
                dst[(size_t)r * C_OUT] = acc[mt][r];
        } else {
#pragma unroll
            for (int r = 0; r < 8; ++r) {
                int row = rowbase + r + mofs;
                if (row < n) H[(size_t)row * C_OUT + col] = acc[mt][r];
            }
        }
    }
}

// ---------------------------------------------------------------------------
// Kernel 3: COO SpMM scatter. One wave per edge: 32 lanes x float4 = 128
// channels. Edge scalars are wave-uniform -> scalarized by the compiler.
// h gather is a coalesced b128 load (h is L2-resident: 51.2MB < 192MB L2).
// Accumulate with relaxed agent-scope f32 atomics (global_atomic_add_f32).
// ---------------------------------------------------------------------------
__global__ __launch_bounds__(256) void gcn_spmm_scatter_kernel(
    const float* __restrict__ H,        // [n, 128]
    const int*   __restrict__ adj_row,  // [e]
    const int*   __restrict__ adj_col,  // [e]
    const float* __restrict__ adj_vals, // [e]
    float* __restrict__ out,            // [n, 128] accumulator
    int e) {
    long long idx = (long long)blockIdx.x * 256 + threadIdx.x;
    int eidx = (int)(idx >> 5);          // one edge per wave
    if (eidx >= e) return;
    int c4 = (int)(idx & 31) << 2;       // this lane's 4 channels

    int   r  = adj_row[eidx];
    int   cl = adj_col[eidx];
    float v  = adj_vals[eidx];

    float4 hv = reinterpret_cast<const float4*>(H + (size_t)cl * C_OUT)[c4 >> 2];
    float* dst = out + (size_t)r * C_OUT + c4;
    __hip_atomic_fetch_add(dst + 0, v * hv.x, __ATOMIC_RELAXED, __HIP_MEMORY_SCOPE_AGENT);
    __hip_atomic_fetch_add(dst + 1, v * hv.y, __ATOMIC_RELAXED, __HIP_MEMORY_SCOPE_AGENT);
    __hip_atomic_fetch_add(dst + 2, v * hv.z, __ATOMIC_RELAXED, __HIP_MEMORY_SCOPE_AGENT);
    __hip_atomic_fetch_add(dst + 3, v * hv.w, __ATOMIC_RELAXED, __HIP_MEMORY_SCOPE_AGENT);
}

// ---------------------------------------------------------------------------
// Kernel 4: out = selu(h * skip + agg + bias), in place on out (== agg)
// ---------------------------------------------------------------------------
__global__ __launch_bounds__(256) void gcn_finalize_kernel(
    const float* __restrict__ H,
    const float* __restrict__ skip,     // [128]
    const float* __restrict__ bias,     // [128]
    float* __restrict__ out,            // [n, 128] holds agg on entry
    int total) {
    int idx = blockIdx.x * 256 + threadIdx.x;
    if (idx >= total) return;
    int c = idx & (C_OUT - 1);
    float x = fmaf(H[idx], skip[c], out[idx]) + bias[c];
    const float alpha = 1.6732632423543772848f;
    const float scale = 1.0507009873554804934f;
    out[idx] = scale * (x > 0.0f ? x : alpha * (__expf(x) - 1.0f));
}

// ---------------------------------------------------------------------------
extern "C" void kernel_launch(void* const* d_in, const int* in_sizes, int n_in,
                              void* d_out, int out_size, void* d_ws, size_t ws_size,
                              hipStream_t stream) {
    const float* X        = (const float*)d_in[0];  // features [n,256]
    const int*   adj_row  = (const int*)  d_in[1];  // [e]
    const int*   adj_col  = (const int*)  d_in[2];  // [e]
    const float* adj_vals = (const float*)d_in[3];  // [e]
    const float* W        = (const float*)d_in[4];  // [256,128]
    const float* bias     = (const float*)d_in[5];  // [128]
    const float* skip     = (const float*)d_in[6];  // [128]
    float*       out      = (float*)d_out;          // [n,128]
    float*       H        = (float*)d_ws;           // scratch: n*128 floats

    const int e = in_sizes[1];
    const int n = in_sizes[0] / D_IN;
    const int total = n * C_OUT;

    // 1) zero accumulator (d_out is poisoned by harness)
    gcn_zero_kernel<<<(total + 255) / 256, 256, 0, stream>>>(out, total);

    // 2) h = X @ W  (WMMA f32, 64 rows/block)
    gcn_gemm_wmma_kernel<<<(n + ROWS_PER_BLOCK - 1) / ROWS_PER_BLOCK, 256, 0,
                          stream>>>(X, W, H, n);

    // 3) agg += A * h  (one wave per edge, f32 atomics into out)
    long long work = (long long)e * 32;
    gcn_spmm_scatter_kernel<<<(unsigned)((work + 255) / 256), 256, 0, stream>>>(
        H, adj_row, adj_col, adj_vals, out, e);

    // 4) out = selu(h*skip + agg + bias)
    gcn_finalize_kernel<<<(total + 255) / 256, 256, 0, stream>>>(
        H, skip, bias, out, total);
}